// TopKModel_47124381172284
// MI455X (gfx1250) — compile-verified
//
#include <hip/hip_runtime.h>
#include <hip/hip_bf16.h>
#include <math.h>

#define K    8
#define TPB  256
#define NBUF 4          // async pipeline depth (tiles in flight = NBUF-1 .. NBUF)

// Insert x (with index xi) into a descending sorted 8-list held in registers.
// Caller guarantees x > v[K-1].
__device__ __forceinline__ void insert8(float (&v)[K], int (&id)[K], float x, int xi) {
#pragma unroll
    for (int s = K - 1; s > 0; --s) {
        if (x > v[s - 1]) { v[s] = v[s - 1]; id[s] = id[s - 1]; }
        else               { v[s] = x;        id[s] = xi;        return; }
    }
    v[0] = x; id[0] = xi;
}

__device__ __forceinline__ void issue_async_tile(unsigned lds_slot, unsigned goff,
                                                 unsigned long long base) {
    asm volatile("global_load_async_to_lds_b128 %0, %1, %2 th:TH_LOAD_NT"
                 :: "v"(lds_slot), "v"(goff), "s"(base) : "memory");
}

__global__ __launch_bounds__(TPB) void topk8_rows_kernel(
        const float* __restrict__ x,
        float* __restrict__ out_vals,
        int*   __restrict__ out_idx,
        int cols)
{
    const int row = blockIdx.x;
    const int tid = threadIdx.x;
    const float* __restrict__ xr = x + (size_t)row * (size_t)cols;

    // Per-lane async staging buffers (wave-private slots -> no barriers needed)
    __shared__ float4 stage[NBUF][TPB];    // 16 KB
    __shared__ float  rv[TPB * K];         //  8 KB
    __shared__ int    ri[TPB * K];         //  8 KB

    float v[K]; int id[K];
#pragma unroll
    for (int k = 0; k < K; ++k) { v[k] = -INFINITY; id[k] = 0x7fffffff; }

    const int nq    = cols >> 2;       // number of full float4 groups in the row
    const int nfull = nq / TPB;        // full block-wide iterations
    const unsigned long long base = (unsigned long long)(uintptr_t)xr;

    // LDS byte offset of this lane's slot in buffer 0 (low 32 bits of a generic
    // pointer to a shared object are the LDS offset on gfx1250).
    const unsigned ldslane = (unsigned)(uintptr_t)&stage[0][tid];

    // ---- prologue: fill the pipeline with up to NBUF-1 tiles ----
#pragma unroll
    for (int t = 0; t < NBUF - 1; ++t) {
        if (t < nfull) {
            issue_async_tile(ldslane + (unsigned)t * (TPB * 16u),
                             (unsigned)(t * TPB + tid) * 16u, base);
        }
    }

    // ---- main loop: issue tile q+3, wait for tile q, process tile q ----
    const int qmain = (nfull > (NBUF - 1)) ? (nfull - (NBUF - 1)) : 0;
    for (int q = 0; q < qmain; ++q) {
        // buffer (q+3)&3 was consumed at iteration q-1; ensure that DS read
        // retired before the async engine overwrites it (essentially free).
        asm volatile("s_wait_dscnt 0" ::: "memory");
        const int qn = q + (NBUF - 1);
        issue_async_tile(ldslane + (unsigned)(qn & (NBUF - 1)) * (TPB * 16u),
                         (unsigned)(qn * TPB + tid) * 16u, base);
        // 4 tiles outstanding; async loads complete in order -> cnt<=3 means
        // tile q has fully landed in LDS.
        asm volatile("s_wait_asynccnt 3" ::: "memory");

        float4 d = stage[q & (NBUF - 1)][tid];
        const int c0 = (q * TPB + tid) << 2;

        float m = fmaxf(fmaxf(d.x, d.y), fmaxf(d.z, d.w));
        if (m > v[K - 1]) {
            if (d.x > v[K - 1]) insert8(v, id, d.x, c0 + 0);
            if (d.y > v[K - 1]) insert8(v, id, d.y, c0 + 1);
            if (d.z > v[K - 1]) insert8(v, id, d.z, c0 + 2);
            if (d.w > v[K - 1]) insert8(v, id, d.w, c0 + 3);
        }
    }

    // ---- pipeline drain: last up-to-3 tiles, waits 2/1/0 ----
    for (int q = qmain; q < nfull; ++q) {
        const int remaining = nfull - 1 - q;   // tiles still outstanding after q
        if      (remaining == 2) asm volatile("s_wait_asynccnt 2" ::: "memory");
        else if (remaining == 1) asm volatile("s_wait_asynccnt 1" ::: "memory");
        else                     asm volatile("s_wait_asynccnt 0" ::: "memory");

        float4 d = stage[q & (NBUF - 1)][tid];
        const int c0 = (q * TPB + tid) << 2;

        float m = fmaxf(fmaxf(d.x, d.y), fmaxf(d.z, d.w));
        if (m > v[K - 1]) {
            if (d.x > v[K - 1]) insert8(v, id, d.x, c0 + 0);
            if (d.y > v[K - 1]) insert8(v, id, d.y, c0 + 1);
            if (d.z > v[K - 1]) insert8(v, id, d.z, c0 + 2);
            if (d.w > v[K - 1]) insert8(v, id, d.w, c0 + 3);
        }
    }

    // ---- tail float4 groups (direct NT dword loads; row base only 4B-aligned) ----
    {
        const int q = nfull * TPB + tid;
        if (q < nq) {
            const int c0 = q << 2;
            float a0 = __builtin_nontemporal_load(xr + c0 + 0);
            float a1 = __builtin_nontemporal_load(xr + c0 + 1);
            float a2 = __builtin_nontemporal_load(xr + c0 + 2);
            float a3 = __builtin_nontemporal_load(xr + c0 + 3);
            float m = fmaxf(fmaxf(a0, a1), fmaxf(a2, a3));
            if (m > v[K - 1]) {
                if (a0 > v[K - 1]) insert8(v, id, a0, c0 + 0);
                if (a1 > v[K - 1]) insert8(v, id, a1, c0 + 1);
                if (a2 > v[K - 1]) insert8(v, id, a2, c0 + 2);
                if (a3 > v[K - 1]) insert8(v, id, a3, c0 + 3);
            }
        }
    }
    // ---- scalar remainder columns ----
    {
        const int rem = cols & 3;
        if (tid < rem) {
            const int c = (nq << 2) + tid;
            float a = __builtin_nontemporal_load(xr + c);
            if (a > v[K - 1]) insert8(v, id, a, c);
        }
    }

    // ---- block reduction: 256 sorted-8 lists -> 1, log2 tree of 8-way merges ----
#pragma unroll
    for (int k = 0; k < K; ++k) { rv[tid * K + k] = v[k]; ri[tid * K + k] = id[k]; }

    for (int s = TPB / 2; s >= 1; s >>= 1) {
        __syncthreads();
        if (tid < s) {
            int i = 0, j = 0;
            float c[K]; int ci[K];
#pragma unroll
            for (int k = 0; k < K; ++k) {
                float av = rv[tid * K + i];
                float bv = rv[(tid + s) * K + j];
                bool ta = (av >= bv);
                c[k]  = ta ? av : bv;
                ci[k] = ta ? ri[tid * K + i] : ri[(tid + s) * K + j];
                if (ta) ++i; else ++j;
            }
#pragma unroll
            for (int k = 0; k < K; ++k) { rv[tid * K + k] = c[k]; ri[tid * K + k] = ci[k]; }
        }
    }
    __syncthreads();

    if (tid < K) {
        out_vals[(size_t)row * K + tid] = rv[tid];
        out_idx [(size_t)row * K + tid] = ri[tid];
    }
}

extern "C" void kernel_launch(void* const* d_in, const int* in_sizes, int n_in,
                              void* d_out, int out_size, void* d_ws, size_t ws_size,
                              hipStream_t stream) {
    (void)n_in; (void)d_ws; (void)ws_size;
    const float* x = (const float*)d_in[0];

    // out = [values (rows*K f32) | indices (rows*K i32)] concatenated flat
    const int rows = out_size / (2 * K);            // 4096
    const int cols = in_sizes[0] / rows;            // 50257

    float* out_vals = (float*)d_out;
    int*   out_idx  = (int*)((float*)d_out + (size_t)rows * K);

    topk8_rows_kernel<<<rows, TPB, 0, stream>>>(x, out_vals, out_idx, cols);
}